// MVN_elbo_autolatent_sp_27427661152651
// MI455X (gfx1250) — compile-verified
//
#include <hip/hip_runtime.h>
#include <hip/hip_bf16.h>

#define NUG 1e-6f

typedef float v2f __attribute__((ext_vector_type(2)));
typedef float v8f __attribute__((ext_vector_type(8)));

// ---------------------------------------------------------------------------
// WMMA f32 16x16x4 wrapper (CDNA5 / gfx1250, wave32).
// A fragment: a[v] = A[M = lane%16, K = 2*(lane/16)+v]
// B fragment: b[v] = B[K = 2*(lane/16)+v, N = lane%16]
// C/D:        c[v] = C[M = v + 8*(lane/16), N = lane%16]
// ---------------------------------------------------------------------------
__device__ inline v8f wmma_f32_16x16x4(v2f a, v2f b, v8f c) {
#if __has_builtin(__builtin_amdgcn_wmma_f32_16x16x4_f32)
  return __builtin_amdgcn_wmma_f32_16x16x4_f32(false, a, false, b, (short)0, c,
                                               false, false);
#else
  int lane = threadIdx.x & 31;
  int half = lane >> 4, col = lane & 15;
#pragma unroll
  for (int v = 0; v < 8; ++v) {
    int Mr = v + half * 8;
    float s = c[v];
#pragma unroll
    for (int kk = 0; kk < 4; ++kk) {
      float av = __shfl((kk & 1) ? a[1] : a[0], (kk >> 1) * 16 + Mr);
      float bv = __shfl((kk & 1) ? b[1] : b[0], (kk >> 1) * 16 + col);
      s += av * bv;
    }
    c[v] = s;
  }
  return c;
#endif
}

// ---------------------------------------------------------------------------
// 1) Prescale coordinates by exp(-log-lengthscale) per component; zero Dinv.
// ---------------------------------------------------------------------------
__global__ void prep_kernel(const float* __restrict__ theta,
                            const float* __restrict__ theta0,
                            const float* __restrict__ thetai,
                            const float* __restrict__ Lmb,
                            float* __restrict__ ths, float* __restrict__ th0s,
                            float* __restrict__ this_,
                            float* __restrict__ DinvZero, int n, int n0, int p,
                            int d, int kap) {
  size_t stride = (size_t)gridDim.x * blockDim.x;
  size_t gid = (size_t)blockIdx.x * blockDim.x + threadIdx.x;
  size_t nd = (size_t)n * d, n0d = (size_t)n0 * d, pd = (size_t)p * d;
  for (size_t idx = gid; idx < (size_t)kap * nd; idx += stride) {
    size_t k = idx / nd, r = idx % nd;
    int j = (int)(r % d);
    ths[idx] = theta[r] * __expf(-Lmb[k * (d + 1) + j]);
  }
  for (size_t idx = gid; idx < (size_t)kap * n0d; idx += stride) {
    size_t k = idx / n0d, r = idx % n0d;
    int j = (int)(r % d);
    th0s[idx] = theta0[r] * __expf(-Lmb[k * (d + 1) + j]);
  }
  for (size_t idx = gid; idx < (size_t)kap * pd; idx += stride) {
    size_t k = idx / pd, r = idx % pd;
    int j = (int)(r % d);
    this_[idx] = thetai[r] * __expf(-Lmb[k * (d + 1) + j]);
  }
  for (size_t idx = gid; idx < (size_t)kap * n; idx += stride)
    DinvZero[idx] = 0.0f;
}

__device__ inline float matern_cov(const float* __restrict__ a,
                                   const float* __restrict__ b, int d) {
  float prod = 1.0f, ss = 0.0f;
  for (int t = 0; t < d; ++t) {
    float s = fabsf(a[t] - b[t]);
    prod *= (1.0f + s);
    ss += s;
  }
  return prod * __expf(-ss);
}

// ---------------------------------------------------------------------------
// 2) c_fi[k] = cov(theta, thetai)   [n, p] per component
// ---------------------------------------------------------------------------
__global__ void cfi_kernel(const float* __restrict__ ths,
                           const float* __restrict__ this_,
                           const float* __restrict__ Lmb,
                           float* __restrict__ cfi, int n, int p, int d) {
  int k = blockIdx.y;
  size_t idx = (size_t)blockIdx.x * blockDim.x + threadIdx.x;
  if (idx >= (size_t)n * p) return;
  int i = (int)(idx / p), j = (int)(idx % p);
  float alpha = __expf(Lmb[(size_t)k * (d + 1) + d]);
  const float* a = ths + ((size_t)k * n + i) * d;
  const float* b = this_ + ((size_t)k * p + j) * d;
  cfi[(size_t)k * n * p + idx] = alpha * matern_cov(a, b, d);
}

// ---------------------------------------------------------------------------
// 3) Ci[k] = cov(thetai, thetai) + NUG*alpha*I
// ---------------------------------------------------------------------------
__global__ void buildCi_kernel(const float* __restrict__ this_,
                               const float* __restrict__ Lmb,
                               float* __restrict__ Ci, int p, int d) {
  int k = blockIdx.y;
  int idx = blockIdx.x * blockDim.x + threadIdx.x;
  if (idx >= p * p) return;
  int a = idx / p, b = idx % p;
  float alpha = __expf(Lmb[(size_t)k * (d + 1) + d]);
  const float* xa = this_ + ((size_t)k * p + a) * d;
  const float* xb = this_ + ((size_t)k * p + b) * d;
  float v = alpha * matern_cov(xa, xb, d);
  if (a == b) v += NUG * alpha;
  Ci[(size_t)k * p * p + idx] = v;
}

// ---------------------------------------------------------------------------
// 4) SPD inverse via in-place Cholesky + triangular inverse; one block per k.
// ---------------------------------------------------------------------------
__global__ void cholinv_kernel(float* __restrict__ Abuf, float* __restrict__ Ybuf,
                               float* __restrict__ Obuf, int p) {
  const int tid = threadIdx.x, nt = blockDim.x;
  float* Aw = Abuf + (size_t)blockIdx.x * p * p;
  float* Y = Ybuf + (size_t)blockIdx.x * p * p;
  float* Out = Obuf + (size_t)blockIdx.x * p * p;
  for (int j = 0; j < p; ++j) {
    __syncthreads();
    float diag = Aw[(size_t)j * p + j];
    float Ljj = sqrtf(fmaxf(diag, 1e-30f));
    float inv = 1.0f / Ljj;
    if (tid == 0) Aw[(size_t)j * p + j] = Ljj;
    for (int i = j + 1 + tid; i < p; i += nt) Aw[(size_t)i * p + j] *= inv;
    __syncthreads();
    for (int i = j + 1 + tid; i < p; i += nt) {
      float Lij = Aw[(size_t)i * p + j];
      for (int t = j + 1; t <= i; ++t)
        Aw[(size_t)i * p + t] -= Lij * Aw[(size_t)t * p + j];
    }
  }
  __syncthreads();
  for (int j = tid; j < p; j += nt) {
    for (int i = 0; i < j; ++i) Y[(size_t)i * p + j] = 0.0f;
    Y[(size_t)j * p + j] = 1.0f / Aw[(size_t)j * p + j];
    for (int i = j + 1; i < p; ++i) {
      float s = 0.0f;
      for (int t = j; t < i; ++t) s += Aw[(size_t)i * p + t] * Y[(size_t)t * p + j];
      Y[(size_t)i * p + j] = -s / Aw[(size_t)i * p + i];
    }
  }
  __syncthreads();
  for (int e = tid; e < p * p; e += nt) {
    int a = e / p, b = e % p;
    int st = a > b ? a : b;
    float s = 0.0f;
    for (int i = st; i < p; ++i) s += Y[(size_t)i * p + a] * Y[(size_t)i * p + b];
    Out[e] = s;
  }
}

// ---------------------------------------------------------------------------
// 5) GEMM1 (WMMA f32 16x16x4): G = c_fi @ Ciinv; fused epilogue accumulates
//    rowsum(G .* c_fi) into DinvRaw via shuffle-reduce + fp32 atomics.
//    Ciinv panel staged in LDS (zero-padded); A loads unconditional+clamped,
//    masked via multiply (no EXEC divergence in the hot loop).
// ---------------------------------------------------------------------------
#define CH1 64
__global__ void gemm1_dinv_kernel(const float* __restrict__ cfi,
                                  const float* __restrict__ Ciinv,
                                  float* __restrict__ DinvRaw, int n, int p) {
  __shared__ float lB[CH1 * 16];
  const int k = blockIdx.z;
  const int lane = threadIdx.x & 31;
  const int wave = threadIdx.x >> 5;
  const int mtile = blockIdx.x * 8 + wave;
  const float* A = cfi + (size_t)k * n * p;    // [n, p] row-major
  const float* Bm = Ciinv + (size_t)k * p * p; // [p, p]
  const int mrow = mtile * 16 + (lane & 15);
  const int mrowc = mrow < n ? mrow : 0;       // clamped (mask handles OOB)
  const float amask = mrow < n ? 1.0f : 0.0f;
  const int ncol0 = blockIdx.y * 16;
  const int ncol = ncol0 + (lane & 15);
  const int koff = (lane >> 4) * 2;
  const float* Arow = A + (size_t)mrowc * p;
  v8f acc = {};
  for (int kk = 0; kk < p; kk += CH1) {
    __syncthreads();
    // cooperative zero-padded load of Ciinv chunk [kk,kk+CH1) x [ncol0,+16)
    for (int e = threadIdx.x; e < CH1 * 16; e += 256) {
      int ki = kk + (e >> 4);
      int c = ncol0 + (e & 15);
      int kic = ki < p ? ki : 0;
      int cc = c < p ? c : 0;
      float msk = (ki < p && c < p) ? 1.0f : 0.0f;
      lB[e] = Bm[(size_t)kic * p + cc] * msk;
    }
    __syncthreads();
#pragma unroll 4
    for (int kl = 0; kl < CH1; kl += 4) {
      v2f a, b;
#pragma unroll
      for (int v = 0; v < 2; ++v) {
        int kil = kl + koff + v;
        a[v] = amask * Arow[kk + kil];      // garbage*0 beyond p (b is 0 there)
        b[v] = lB[kil * 16 + (lane & 15)];
      }
      acc = wmma_f32_16x16x4(a, b, acc);
    }
  }
  const int half = lane >> 4;
  const int ncolc = ncol < p ? ncol : 0;
#pragma unroll
  for (int v = 0; v < 8; ++v) {
    int Mr = mtile * 16 + v + half * 8;
    int Mrc = Mr < n ? Mr : 0;
    float msk = (Mr < n && ncol < p) ? 1.0f : 0.0f;
    float val = acc[v] * A[(size_t)Mrc * p + ncolc] * msk;
    val += __shfl_xor(val, 1);
    val += __shfl_xor(val, 2);
    val += __shfl_xor(val, 4);
    val += __shfl_xor(val, 8);
    if ((lane & 15) == 0 && Mr < n)
      atomicAdd(&DinvRaw[(size_t)k * n + Mr], val);
  }
}

// ---------------------------------------------------------------------------
// 6) Dinv = 1 / max(alpha*(1+NUG) - rowsum, NUG), in place.
// ---------------------------------------------------------------------------
__global__ void dinv_kernel(float* __restrict__ Dinv,
                            const float* __restrict__ Lmb, int n, int d,
                            int kap) {
  size_t idx = (size_t)blockIdx.x * blockDim.x + threadIdx.x;
  if (idx >= (size_t)kap * n) return;
  int k = (int)(idx / n);
  float alpha = __expf(Lmb[(size_t)k * (d + 1) + d]);
  float delta = fmaxf(alpha * (1.0f + NUG) - Dinv[idx], NUG);
  Dinv[idx] = 1.0f / delta;
}

// ---------------------------------------------------------------------------
// 7) GEMM2 (WMMA f32 16x16x4): P = c_fi^T @ (Dinv .* c_fi). Both 16-column
//    K-panels staged in LDS in 128-row chunks (zero-padded, Dinv pre-applied);
//    8 waves split each chunk's K; LDS cross-wave reduce at the end.
// ---------------------------------------------------------------------------
#define CH2 128
__global__ void gemm2_P_kernel(const float* __restrict__ cfi,
                               const float* __restrict__ Dinv,
                               float* __restrict__ Pbuf, int n, int p) {
  __shared__ float lA[CH2 * 16];
  __shared__ float lB[CH2 * 16];
  __shared__ float red[8 * 256];
  const int k = blockIdx.z;
  const int lane = threadIdx.x & 31;
  const int wave = threadIdx.x >> 5;
  const float* A = cfi + (size_t)k * n * p;
  const float* Dv = Dinv + (size_t)k * n;
  const int jm0 = blockIdx.x * 16;
  const int jn0 = blockIdx.y * 16;
  const int koff = (lane >> 4) * 2;
  v8f acc = {};
  for (int kk = 0; kk < n; kk += CH2) {
    __syncthreads();
    for (int e = threadIdx.x; e < CH2 * 16; e += 256) {
      int ki = kk + (e >> 4);
      int kic = ki < n ? ki : 0;
      int cA = jm0 + (e & 15);
      int cB = jn0 + (e & 15);
      int cAc = cA < p ? cA : 0;
      int cBc = cB < p ? cB : 0;
      float mA = (ki < n && cA < p) ? 1.0f : 0.0f;
      float mB = (ki < n && cB < p) ? 1.0f : 0.0f;
      const float* Rk = A + (size_t)kic * p;
      lA[e] = Rk[cAc] * mA;
      lB[e] = Dv[kic] * Rk[cBc] * mB;
      // prefetch next chunk's A panel row (clamped, unconditional)
      int pki = (ki + CH2 < n) ? ki + CH2 : kic;
      __builtin_prefetch(A + (size_t)pki * p + cAc, 0, 1);
    }
    __syncthreads();
    const int base = wave * (CH2 / 8);
#pragma unroll
    for (int kl = 0; kl < CH2 / 8; kl += 4) {
      v2f a, b;
#pragma unroll
      for (int v = 0; v < 2; ++v) {
        int kil = base + kl + koff + v;
        a[v] = lA[kil * 16 + (lane & 15)];
        b[v] = lB[kil * 16 + (lane & 15)];
      }
      acc = wmma_f32_16x16x4(a, b, acc);
    }
  }
  __syncthreads();
#pragma unroll
  for (int v = 0; v < 8; ++v) red[wave * 256 + v * 32 + lane] = acc[v];
  __syncthreads();
  int v = threadIdx.x >> 5, l2 = threadIdx.x & 31;
  float s = 0.0f;
#pragma unroll
  for (int w = 0; w < 8; ++w) s += red[w * 256 + v * 32 + l2];
  int Mr = blockIdx.x * 16 + v + (l2 >> 4) * 8;
  int Nc = blockIdx.y * 16 + (l2 & 15);
  if (Mr < p && Nc < p)
    Pbuf[(size_t)k * p * p + (size_t)Mr * p + Nc] = s;
}

// ---------------------------------------------------------------------------
// 8) B = Ci + P  (Ci recomputed, cheap) -> written into matA for cholinv.
// ---------------------------------------------------------------------------
__global__ void buildB_kernel(const float* __restrict__ this_,
                              const float* __restrict__ Lmb,
                              const float* __restrict__ Pbuf,
                              float* __restrict__ Bout, int p, int d) {
  int k = blockIdx.y;
  int idx = blockIdx.x * blockDim.x + threadIdx.x;
  if (idx >= p * p) return;
  int a = idx / p, b = idx % p;
  float alpha = __expf(Lmb[(size_t)k * (d + 1) + d]);
  const float* xa = this_ + ((size_t)k * p + a) * d;
  const float* xb = this_ + ((size_t)k * p + b) * d;
  float v = alpha * matern_cov(xa, xb, d);
  if (a == b) v += NUG * alpha;
  Bout[(size_t)k * p * p + idx] = v + Pbuf[(size_t)k * p * p + idx];
}

// ---------------------------------------------------------------------------
// 9) t[k][j] = sum_i c_fi[i,j] * Dinv[i] * g[i]
// ---------------------------------------------------------------------------
__global__ void tvec_kernel(const float* __restrict__ cfi,
                            const float* __restrict__ Dinv,
                            const float* __restrict__ g,
                            float* __restrict__ tvec, int n, int p) {
  __shared__ float red[256];
  int j = blockIdx.x, k = blockIdx.y, tid = threadIdx.x;
  const float* A = cfi + (size_t)k * n * p;
  const float* Dv = Dinv + (size_t)k * n;
  const float* gk = g + (size_t)k * n;
  float acc = 0.0f;
  for (int i = tid; i < n; i += 256) acc += A[(size_t)i * p + j] * Dv[i] * gk[i];
  red[tid] = acc;
  __syncthreads();
  for (int s = 128; s > 0; s >>= 1) {
    if (tid < s) red[tid] += red[tid + s];
    __syncthreads();
  }
  if (tid == 0) tvec[(size_t)k * p + j] = red[0];
}

// 10) s = Binv @ t
__global__ void svec_kernel(const float* __restrict__ Binv,
                            const float* __restrict__ tvec,
                            float* __restrict__ svec, int p) {
  int k = blockIdx.x;
  const float* B = Binv + (size_t)k * p * p;
  const float* t = tvec + (size_t)k * p;
  for (int j = threadIdx.x; j < p; j += blockDim.x) {
    float s = 0.0f;
    for (int u = 0; u < p; ++u) s += B[(size_t)j * p + u] * t[u];
    svec[(size_t)k * p + j] = s;
  }
}

// 11) Cinv_g[i] = Dinv[i] * (g[i] - c_fi[i,:] @ s)
__global__ void cinvg_kernel(const float* __restrict__ cfi,
                             const float* __restrict__ Dinv,
                             const float* __restrict__ g,
                             const float* __restrict__ svec,
                             float* __restrict__ cinvg, int n, int p, int kap) {
  size_t idx = (size_t)blockIdx.x * blockDim.x + threadIdx.x;
  if (idx >= (size_t)kap * n) return;
  int k = (int)(idx / n), i = (int)(idx % n);
  const float* row = cfi + ((size_t)k * n + i) * p;
  const float* s = svec + (size_t)k * p;
  float dot = 0.0f;
  for (int j = 0; j < p; ++j) dot += row[j] * s[j];
  cinvg[idx] = Dinv[idx] * (g[idx] - dot);
}

// 12) ghat[k][i0] = sum_i cov(theta0[i0], theta[i]) * Cinv_g[i]
__global__ void ghat_kernel(const float* __restrict__ th0s,
                            const float* __restrict__ ths,
                            const float* __restrict__ cinvg,
                            const float* __restrict__ Lmb,
                            float* __restrict__ ghat, int n, int n0, int d) {
  __shared__ float red[256];
  int i0 = blockIdx.x, k = blockIdx.y, tid = threadIdx.x;
  float alpha = __expf(Lmb[(size_t)k * (d + 1) + d]);
  const float* x0 = th0s + ((size_t)k * n0 + i0) * d;
  float x0r[16];
  for (int j = 0; j < d; ++j) x0r[j] = x0[j];
  const float* X = ths + (size_t)k * n * d;
  const float* cg = cinvg + (size_t)k * n;
  float acc = 0.0f;
  for (int i = tid; i < n; i += 256) {
    const float* xi = X + (size_t)i * d;
    float prod = 1.0f, ss = 0.0f;
    for (int j = 0; j < d; ++j) {
      float s = fabsf(x0r[j] - xi[j]);
      prod *= (1.0f + s);
      ss += s;
    }
    acc += prod * __expf(-ss) * cg[i];
  }
  red[tid] = acc;
  __syncthreads();
  for (int s = 128; s > 0; s >>= 1) {
    if (tid < s) red[tid] += red[tid + s];
    __syncthreads();
  }
  if (tid == 0) ghat[(size_t)k * n0 + i0] = alpha * red[0];
}

// 13) out[row,col] = Fstd[row] * sum_k Phi[row,k]*ghat[k,col] + Fmean[row]
__global__ void fhat_kernel(const float* __restrict__ Phi,
                            const float* __restrict__ ghat,
                            const float* __restrict__ Fmean,
                            const float* __restrict__ Fstd,
                            float* __restrict__ out, int m, int n0, int kap) {
  size_t idx = (size_t)blockIdx.x * blockDim.x + threadIdx.x;
  if (idx >= (size_t)m * n0) return;
  int row = (int)(idx / n0), col = (int)(idx % n0);
  float s = 0.0f;
  for (int k = 0; k < kap; ++k)
    s += Phi[(size_t)row * kap + k] * ghat[(size_t)k * n0 + col];
  out[idx] = s * Fstd[row] + Fmean[row];
}

// ---------------------------------------------------------------------------
extern "C" void kernel_launch(void* const* d_in, const int* in_sizes, int n_in,
                              void* d_out, int out_size, void* d_ws,
                              size_t ws_size, hipStream_t stream) {
  const float* theta0 = (const float*)d_in[0];
  const float* Lmb = (const float*)d_in[1];
  // d_in[2] = lsigma2 (unused by the reference computation)
  const float* Phi = (const float*)d_in[3];
  const float* theta = (const float*)d_in[4];
  const float* thetai = (const float*)d_in[5];
  const float* Mg = (const float*)d_in[6];
  const float* Fmean = (const float*)d_in[7];
  const float* Fstd = (const float*)d_in[8];
  float* out = (float*)d_out;
  (void)n_in; (void)out_size; (void)ws_size;

  const int m = in_sizes[7];
  const int kap = in_sizes[3] / m;
  const int n = in_sizes[6] / kap;
  const int d = in_sizes[4] / n;
  const int p = in_sizes[5] / d;
  const int n0 = in_sizes[0] / d;

  float* w = (float*)d_ws;
  size_t off = 0;
  auto alloc = [&](size_t cnt) { float* ptr = w + off; off += cnt; return ptr; };
  float* ths = alloc((size_t)kap * n * d);
  float* th0s = alloc((size_t)kap * n0 * d);
  float* this_ = alloc((size_t)kap * p * d);
  float* cfi = alloc((size_t)kap * n * p);
  float* matA = alloc((size_t)kap * p * p);   // Ci, then B (both factored here)
  float* Ybuf = alloc((size_t)kap * p * p);
  float* invB = alloc((size_t)kap * p * p);   // Ciinv, then Binv
  float* Pbuf = alloc((size_t)kap * p * p);
  float* Dinv = alloc((size_t)kap * n);
  float* tvec = alloc((size_t)kap * p);
  float* svec = alloc((size_t)kap * p);
  float* cinvg = alloc((size_t)kap * n);
  float* ghat = alloc((size_t)kap * n0);

  auto cdiv = [](size_t a, size_t b) { return (unsigned)((a + b - 1) / b); };

  prep_kernel<<<2048, 256, 0, stream>>>(theta, theta0, thetai, Lmb, ths, th0s,
                                        this_, Dinv, n, n0, p, d, kap);
  cfi_kernel<<<dim3(cdiv((size_t)n * p, 256), kap), 256, 0, stream>>>(
      ths, this_, Lmb, cfi, n, p, d);
  buildCi_kernel<<<dim3(cdiv((size_t)p * p, 256), kap), 256, 0, stream>>>(
      this_, Lmb, matA, p, d);
  cholinv_kernel<<<kap, 256, 0, stream>>>(matA, Ybuf, invB, p);
  gemm1_dinv_kernel<<<dim3(cdiv(cdiv(n, 16), 8), cdiv(p, 16), kap), 256, 0,
                      stream>>>(cfi, invB, Dinv, n, p);
  dinv_kernel<<<cdiv((size_t)kap * n, 256), 256, 0, stream>>>(Dinv, Lmb, n, d,
                                                              kap);
  gemm2_P_kernel<<<dim3(cdiv(p, 16), cdiv(p, 16), kap), 256, 0, stream>>>(
      cfi, Dinv, Pbuf, n, p);
  buildB_kernel<<<dim3(cdiv((size_t)p * p, 256), kap), 256, 0, stream>>>(
      this_, Lmb, Pbuf, matA, p, d);
  cholinv_kernel<<<kap, 256, 0, stream>>>(matA, Ybuf, invB, p);
  tvec_kernel<<<dim3(p, kap), 256, 0, stream>>>(cfi, Dinv, Mg, tvec, n, p);
  svec_kernel<<<kap, 256, 0, stream>>>(invB, tvec, svec, p);
  cinvg_kernel<<<cdiv((size_t)kap * n, 256), 256, 0, stream>>>(
      cfi, Dinv, Mg, svec, cinvg, n, p, kap);
  ghat_kernel<<<dim3(n0, kap), 256, 0, stream>>>(th0s, ths, cinvg, Lmb, ghat,
                                                 n, n0, d);
  fhat_kernel<<<cdiv((size_t)m * n0, 256), 256, 0, stream>>>(
      Phi, ghat, Fmean, Fstd, out, m, n0, kap);
}